// SwapTestCircuit_91199335563793
// MI455X (gfx1250) — compile-verified
//
#include <hip/hip_runtime.h>
#include <math.h>

typedef float f4 __attribute__((ext_vector_type(4)));
typedef int   v4i __attribute__((ext_vector_type(4)));

typedef __attribute__((address_space(1))) v4i* gptr_v4i;  // global
typedef __attribute__((address_space(3))) v4i* lptr_v4i;  // LDS

#define BLOCK 256

#if defined(__HIP_DEVICE_COMPILE__) && \
    __has_builtin(__builtin_amdgcn_global_load_async_to_lds_b128) && \
    __has_builtin(__builtin_amdgcn_s_wait_asynccnt)
#define HAVE_ASYNC_LDS 1
#else
#define HAVE_ASYNC_LDS 0
#endif

__device__ __forceinline__ float fast_tanh(float x) {
#if defined(__HIP_DEVICE_COMPILE__) && __has_builtin(__builtin_amdgcn_tanhf)
  return __builtin_amdgcn_tanhf(x);   // v_tanh_f32 (gfx1250 tanh-insts)
#else
  return tanhf(x);
#endif
}

// cos of (x revolutions) == cos(2*pi*x); valid without range reduction for |x| <= 0.5
__device__ __forceinline__ float cos_rev(float x) {
#if defined(__HIP_DEVICE_COMPILE__) && __has_builtin(__builtin_amdgcn_cosf)
  return __builtin_amdgcn_cosf(x);    // v_cos_f32 (argument in revolutions)
#else
  return cosf(x * 6.28318530717958647692f);
#endif
}

// psi/phi are (nrows, 8) f32; we consume columns 0..3 of each row.
// out[i] = 0.5 * (1 + (prod_j cos((tanh(psi_ij) - tanh(phi_ij)) * pi/2))^2)
__global__ __launch_bounds__(BLOCK) void SwapTestCircuit_kernel(
    const f4* __restrict__ psi, const f4* __restrict__ phi,
    float* __restrict__ out, int nrows)
{
  __shared__ f4 lds_psi[BLOCK];
  __shared__ f4 lds_phi[BLOCK];

  const int tid = threadIdx.x;
  const int row = blockIdx.x * BLOCK + tid;
  // Clamp instead of early-return so barriers/async ops see full EXEC.
  const int r = row < nrows ? row : (nrows - 1);
  const size_t idx = 2ull * (size_t)r;   // row stride = 8 floats = 2 x f4; take first f4

#if HAVE_ASYNC_LDS
  // CDNA5 async global -> LDS copy (tracked by ASYNCcnt), 16B per lane.
  __builtin_amdgcn_global_load_async_to_lds_b128(
      (gptr_v4i)(psi + idx), (lptr_v4i)&lds_psi[tid], 0, 0);
  __builtin_amdgcn_global_load_async_to_lds_b128(
      (gptr_v4i)(phi + idx), (lptr_v4i)&lds_phi[tid], 0, 0);
  __builtin_amdgcn_s_wait_asynccnt(0);
  __syncthreads();
  const f4 a = lds_psi[tid];
  const f4 b = lds_phi[tid];
#else
  const f4 a = psi[idx];
  const f4 b = phi[idx];
#endif

  // theta = tanh(x)*pi; cos((a-b)/2) in revolutions = (tanh_a - tanh_b)/4,
  // which lies in (-0.5, 0.5): v_cos_f32 needs no range reduction.
  const float d0 = (fast_tanh(a.x) - fast_tanh(b.x)) * 0.25f;
  const float d1 = (fast_tanh(a.y) - fast_tanh(b.y)) * 0.25f;
  const float d2 = (fast_tanh(a.z) - fast_tanh(b.z)) * 0.25f;
  const float d3 = (fast_tanh(a.w) - fast_tanh(b.w)) * 0.25f;

  const float overlap = cos_rev(d0) * cos_rev(d1) * cos_rev(d2) * cos_rev(d3);
  const float result = fmaf(overlap * overlap, 0.5f, 0.5f);

  if (row < nrows) out[row] = result;
}

extern "C" void kernel_launch(void* const* d_in, const int* in_sizes, int n_in,
                              void* d_out, int out_size, void* d_ws, size_t ws_size,
                              hipStream_t stream) {
  (void)n_in; (void)d_ws; (void)ws_size;
  const f4* psi = (const f4*)d_in[0];
  const f4* phi = (const f4*)d_in[1];
  float* out = (float*)d_out;
  const int nrows = in_sizes[0] / 8;   // (B, 8) f32
  (void)out_size;

  const int grid = (nrows + BLOCK - 1) / BLOCK;
  SwapTestCircuit_kernel<<<grid, BLOCK, 0, stream>>>(psi, phi, out, nrows);
}